// AttrDec_44135083933972
// MI455X (gfx1250) — compile-verified
//
#include <hip/hip_runtime.h>

typedef float v2f __attribute__((ext_vector_type(2)));
typedef float v8f __attribute__((ext_vector_type(8)));

#define FO 128  // output feature width of both layers

// ---------------- degree histogram / normalization ----------------
__global__ __launch_bounds__(256) void k_zero_i32(int* __restrict__ p, int n) {
  int i = blockIdx.x * blockDim.x + threadIdx.x;
  if (i < n) p[i] = 0;
}

__global__ __launch_bounds__(256) void k_count(const int* __restrict__ dst,
                                               int* __restrict__ cnt, int e) {
  int i = blockIdx.x * blockDim.x + threadIdx.x;
  if (i < e) atomicAdd(&cnt[dst[i]], 1);
}

__global__ __launch_bounds__(256) void k_dinv(const int* __restrict__ cnt,
                                              float* __restrict__ dinv, int n) {
  int i = blockIdx.x * blockDim.x + threadIdx.x;
  if (i < n) dinv[i] = rsqrtf((float)cnt[i] + 1.0f);  // +1 = self-loop
}

// ---------------- hierarchical exclusive scan (N <= 256*256) ----------------
__global__ __launch_bounds__(256) void k_scan1(const int* __restrict__ cnt,
                                               int* __restrict__ rowstart,
                                               int* __restrict__ sums, int n) {
  __shared__ int s[256];
  int i = blockIdx.x * 256 + threadIdx.x;
  int v = (i < n) ? cnt[i] : 0;
  s[threadIdx.x] = v;
  __syncthreads();
  for (int off = 1; off < 256; off <<= 1) {
    int tmp = 0;
    if (threadIdx.x >= off) tmp = s[threadIdx.x - off];
    __syncthreads();
    if (threadIdx.x >= off) s[threadIdx.x] += tmp;
    __syncthreads();
  }
  if (i < n) rowstart[i] = s[threadIdx.x] - v;  // exclusive within chunk
  if (threadIdx.x == 255) sums[blockIdx.x] = s[255];
}

__global__ __launch_bounds__(256) void k_scan2(int* __restrict__ sums, int n) {
  __shared__ int s[256];
  int v = (threadIdx.x < n) ? sums[threadIdx.x] : 0;
  s[threadIdx.x] = v;
  __syncthreads();
  for (int off = 1; off < 256; off <<= 1) {
    int tmp = 0;
    if (threadIdx.x >= off) tmp = s[threadIdx.x - off];
    __syncthreads();
    if (threadIdx.x >= off) s[threadIdx.x] += tmp;
    __syncthreads();
  }
  if (threadIdx.x < n) sums[threadIdx.x] = s[threadIdx.x] - v;  // exclusive
}

__global__ __launch_bounds__(256) void k_scan3(int* __restrict__ rowstart,
                                               int* __restrict__ cursor,
                                               const int* __restrict__ sums, int n) {
  int i = blockIdx.x * 256 + threadIdx.x;
  if (i < n) {
    int r = rowstart[i] + sums[blockIdx.x];
    rowstart[i] = r;
    cursor[i] = r;
  }
}

// ---------------- CSR fill: bucket (src, weight) by dst ----------------
__global__ __launch_bounds__(256) void k_fill(const int* __restrict__ src,
                                              const int* __restrict__ dst,
                                              const float* __restrict__ dinv,
                                              int* __restrict__ cursor,
                                              int* __restrict__ csr_src,
                                              float* __restrict__ csr_w, int e) {
  int i = blockIdx.x * blockDim.x + threadIdx.x;
  if (i < e) {
    int s = src[i];
    int d = dst[i];
    int pos = atomicAdd(&cursor[d], 1);
    csr_src[pos] = s;
    csr_w[pos] = dinv[s] * dinv[d];  // precomputed edge norm
  }
}

// ---------------- fp32 WMMA GEMM: C[M x 128] = act(A + bias)[M x K] * B[K x 128]
// One block = 8 waves, 16 rows x 128 cols. A tile staged in LDS (padded stride)
// with optional fused bias+ReLU. M must be a multiple of 16 (50000 = 3125*16).
__global__ __launch_bounds__(256) void k_gemm_wmma(const float* __restrict__ A,
                                                   const float* __restrict__ B,
                                                   float* __restrict__ C, int K,
                                                   const float* __restrict__ bias) {
  __shared__ float As[16 * 132];  // max K=128, stride K+4=132
  const int tid  = threadIdx.x;
  const int lane = tid & 31;
  const int wave = tid >> 5;
  const int row0 = blockIdx.x * 16;
  const int KP   = K + 4;
  const bool act = (bias != nullptr);

  // cooperative stage of A tile: 16 rows x K floats, float4 per chunk
  const int chunks = (16 * K) >> 2;
  const int cpr    = K >> 2;
  for (int idx = tid; idx < chunks; idx += 256) {
    int r  = idx / cpr;
    int c4 = (idx - r * cpr) << 2;
    float4 v = *(const float4*)&A[(size_t)(row0 + r) * K + c4];
    if (act) {  // fused relu(A + bias) for layer 2
      const float4 bb = *(const float4*)&bias[c4];
      v.x = fmaxf(v.x + bb.x, 0.0f);
      v.y = fmaxf(v.y + bb.y, 0.0f);
      v.z = fmaxf(v.z + bb.z, 0.0f);
      v.w = fmaxf(v.w + bb.w, 0.0f);
    }
    *(float4*)&As[r * KP + c4] = v;
  }
  __syncthreads();

  // 32-bit A 16x4: lanes0-15 M=lane, VGPR0 = K(k0+2*kh), VGPR1 = next K
  // 32-bit B 4x16: V0 = rows {k0, k0+2} across lane halves, V1 = {k0+1, k0+3}
  const int m  = lane & 15;
  const int kh = lane >> 4;
  const int n  = (wave << 4) + (lane & 15);

  v8f acc = {};
  for (int k0 = 0; k0 < K; k0 += 4) {
    const int ka = k0 + (kh << 1);
    v2f a, b;
    a.x = As[m * KP + ka];
    a.y = As[m * KP + ka + 1];
    b.x = B[(size_t)ka * FO + n];
    b.y = B[(size_t)(ka + 1) * FO + n];
    acc = __builtin_amdgcn_wmma_f32_16x16x4_f32(false, a, false, b,
                                                (short)0, acc, false, false);
  }

#pragma unroll
  for (int r = 0; r < 8; ++r) {
    C[(size_t)(row0 + r + (kh << 3)) * FO + n] = acc[r];
  }
}

// ---------------- atomic-free CSR gather aggregation ----------------
// out[i] = dinv[i]^2 * t[i] + sum_{s in in(i)} w_e * t[s] (+ bias)
// One wave per node; lane owns 4 consecutive features. Edge list is loaded
// cooperatively (one coalesced load per 32 edges) and broadcast lane-to-lane
// via __shfl (wave32 lane-permute path) so only the 512B t[src] gather is on
// the per-edge critical path.
__global__ __launch_bounds__(256) void k_gather(const float* __restrict__ t,
                                                const int* __restrict__ csr_src,
                                                const float* __restrict__ csr_w,
                                                const int* __restrict__ rowstart,
                                                const int* __restrict__ cnt,
                                                const float* __restrict__ dinv,
                                                const float* __restrict__ bias,
                                                float* __restrict__ out, int n) {
  int gid = blockIdx.x * blockDim.x + threadIdx.x;
  int i = gid >> 5;
  if (i >= n) return;
  int lane = threadIdx.x & 31;
  int c = lane << 2;
  float di = dinv[i];
  float w0 = di * di;
  float4 acc = *(const float4*)&t[(size_t)i * FO + c];  // self-loop term
  acc.x *= w0; acc.y *= w0; acc.z *= w0; acc.w *= w0;

  const int beg = rowstart[i];
  const int deg = cnt[i];
  for (int base = 0; base < deg; base += 32) {
    const int m = min(32, deg - base);
    int   sl = 0;
    float wl = 0.0f;
    if (base + lane < deg) {               // coalesced edge-list load
      sl = csr_src[beg + base + lane];
      wl = csr_w[beg + base + lane];
    }
    for (int j = 0; j < m; ++j) {
      int   s = __shfl(sl, j, 32);         // broadcast edge j to all lanes
      float w = __shfl(wl, j, 32);
      float4 v = *(const float4*)&t[(size_t)s * FO + c];
      acc.x += w * v.x;
      acc.y += w * v.y;
      acc.z += w * v.z;
      acc.w += w * v.w;
    }
  }
  if (bias) {
    const float4 bb = *(const float4*)&bias[c];
    acc.x += bb.x; acc.y += bb.y; acc.z += bb.z; acc.w += bb.w;
  }
  *(float4*)&out[(size_t)i * FO + c] = acc;
}

extern "C" void kernel_launch(void* const* d_in, const int* in_sizes, int n_in,
                              void* d_out, int out_size, void* d_ws, size_t ws_size,
                              hipStream_t stream) {
  const float* z  = (const float*)d_in[0];   // [N,64]
  const int*   ei = (const int*)d_in[1];     // [2,E]
  const float* W1 = (const float*)d_in[2];   // [64,128]
  const float* b1 = (const float*)d_in[3];   // [128]
  const float* W2 = (const float*)d_in[4];   // [128,128]
  const float* b2 = (const float*)d_in[5];   // [128]
  float* out = (float*)d_out;                // [N,128]

  const int N = in_sizes[0] / 64;
  const int E = in_sizes[1] / 2;
  const int* src = ei;
  const int* dst = ei + E;

  // workspace layout (16B-aligned regions)
  char* p = (char*)d_ws;
  size_t nPad = (size_t)((N + 3) & ~3);
  size_t ePad = (size_t)((E + 3) & ~3);
  float* dinv     = (float*)p;            p += nPad * 4;
  int*   cnt      = (int*)p;              p += nPad * 4;
  int*   rowstart = (int*)p;              p += nPad * 4;
  int*   cursor   = (int*)p;              p += nPad * 4;
  int*   sums     = (int*)p;              p += 256 * 4;
  int*   csr_src  = (int*)p;              p += ePad * 4;
  float* csr_w    = (float*)p;            p += ePad * 4;
  float* t        = (float*)p;            p += (size_t)N * FO * 4;
  float* h        = (float*)p;

  const int nbN = (N + 255) / 256;        // also #scan chunks (196 <= 256)
  const int nbE = (E + 255) / 256;
  const int mtiles = N / 16;              // 3125 exact
  const int nodeWaveBlocks = (int)(((size_t)N * 32 + 255) / 256);

  // build dst-CSR + normalization (int atomics only)
  k_zero_i32<<<nbN, 256, 0, stream>>>(cnt, N);
  k_count   <<<nbE, 256, 0, stream>>>(dst, cnt, E);
  k_dinv    <<<nbN, 256, 0, stream>>>(cnt, dinv, N);
  k_scan1   <<<nbN, 256, 0, stream>>>(cnt, rowstart, sums, N);
  k_scan2   <<<1,   256, 0, stream>>>(sums, nbN);
  k_scan3   <<<nbN, 256, 0, stream>>>(rowstart, cursor, sums, N);
  k_fill    <<<nbE, 256, 0, stream>>>(src, dst, dinv, cursor, csr_src, csr_w, E);

  // layer 1: t = z @ W1 ; h = agg(t)               (bias+relu deferred)
  k_gemm_wmma<<<mtiles, 256, 0, stream>>>(z, W1, t, 64, nullptr);
  k_gather<<<nodeWaveBlocks, 256, 0, stream>>>(t, csr_src, csr_w, rowstart, cnt,
                                               dinv, nullptr, h, N);

  // layer 2: t = relu(h + b1) @ W2 ; out = agg(t) + b2
  k_gemm_wmma<<<mtiles, 256, 0, stream>>>(h, W2, t, 128, b1);
  k_gather<<<nodeWaveBlocks, 256, 0, stream>>>(t, csr_src, csr_w, rowstart, cnt,
                                               dinv, b2, out, N);
}